// LinearElementRef_78323023610112
// MI455X (gfx1250) — compile-verified
//
#include <hip/hip_runtime.h>

#define N_NODES 20000
#define N_ROWS  16
#define N_CH    128
#define OUT_DIM 128
#define N_ELEM  10
#define PATH_W  0.08838834764831845f   // 1/sqrt(128)
#define M_TILE  64

typedef __attribute__((ext_vector_type(2))) float v2f;
typedef __attribute__((ext_vector_type(8))) float v8f;

__device__ __forceinline__ int l_of_row(int r) {
    // rows 0 -> l=0, 1..3 -> l=1, 4..8 -> l=2, 9..15 -> l=3
    return (r == 0) ? 0 : (r < 4) ? 1 : (r < 9) ? 2 : 3;
}

__device__ __forceinline__ int elem_of(const int* __restrict__ y, int n) {
    int e = 0;
#pragma unroll
    for (int i = 0; i < N_ELEM; ++i)
        if (y[n * N_ELEM + i] != 0) e = i;
    return e;
}

// CDNA5 async global->LDS copy (16B per lane), tracked by ASYNCcnt.
// VDST = per-lane LDS byte address, VADDR = 64-bit global address (GV mode).
__device__ __forceinline__ void async_copy_b128(void* lds_dst, const void* gsrc) {
    unsigned           lds = (unsigned)(unsigned long long)(uintptr_t)lds_dst; // flat->LDS offset (low 32 bits)
    unsigned long long ga  = (unsigned long long)(uintptr_t)gsrc;
    asm volatile("global_load_async_to_lds_b128 %0, %1, off"
                 :: "v"(lds), "v"(ga) : "memory");
}
__device__ __forceinline__ void wait_async_zero() {
    asm volatile("s_wait_asynccnt 0" ::: "memory");
}

// ---- ws layout (ints): [0..9] counts, [16..25] cursors, [32..42] offsets,
// ----                   [64 .. 64+N_NODES) sorted node order
__global__ void zero_ws_kernel(int* ws) {
    int t = threadIdx.x;
    if (t < 48) ws[t] = 0;
}

__global__ void count_kernel(const int* __restrict__ y, int* ws) {
    int n = blockIdx.x * blockDim.x + threadIdx.x;
    if (n < N_NODES) {
        int e = elem_of(y, n);
        atomicAdd(&ws[e], 1);
    }
}

__global__ void scan_kernel(int* ws) {
    if (threadIdx.x == 0) {
        int off = 0;
        for (int e = 0; e < N_ELEM; ++e) {
            ws[32 + e] = off;   // offsets
            ws[16 + e] = off;   // cursors
            off += ws[e];       // counts
        }
        ws[32 + N_ELEM] = off;
    }
}

__global__ void scatter_kernel(const int* __restrict__ y, int* ws) {
    int n = blockIdx.x * blockDim.x + threadIdx.x;
    if (n < N_NODES) {
        int e = elem_of(y, n);
        int pos = atomicAdd(&ws[16 + e], 1);
        ws[64 + pos] = n;
    }
}

// ---- main GEMM: per (element, row): [count_e x 128] @ [128 x 128] ----
// Block: 256 threads = 8 wave32. M tile = 64 sorted nodes (4 sub-tiles),
// wave w owns output columns [w*16, w*16+16). 128 WMMA per wave per block.
// A and B staged into LDS via global_load_async_to_lds_b128.
__global__ __launch_bounds__(256) void gemm_kernel(
    const float* __restrict__ x, const float* __restrict__ w,
    float* __restrict__ out, const int* __restrict__ ws)
{
    const int e  = blockIdx.z;
    const int r  = blockIdx.y;
    const int mt = blockIdx.x;

    const int cnt = ws[e];            // counts
    const int m0  = mt * M_TILE;
    if (m0 >= cnt) return;            // uniform early-exit (EXEC all-1s below)
    const int base = ws[32 + e];      // offsets
    const int* order = ws + 64;

    __shared__ int   nids[M_TILE];
    __shared__ float As[M_TILE][132];   // padded stride 132: conflict-free b64 frags
    __shared__ float Bs[N_CH][132];     // natural layout [c][d]; rows 16B-aligned

    const int tid = threadIdx.x;

    if (tid < M_TILE) {
        int mm = m0 + tid;
        nids[tid] = order[base + (mm < cnt ? mm : cnt - 1)];
    }
    __syncthreads();

    // ---- async stage A: 64 rows x 128 ch, 16B per lane, 8 ops/thread ----
#pragma unroll
    for (int i = tid; i < M_TILE * (N_CH / 4); i += 256) {
        int m  = i >> 5;                      // 32 float4 per row
        int c4 = (i & 31) << 2;
        async_copy_b128(&As[m][c4],
                        x + (size_t)nids[m] * (N_ROWS * N_CH) + (size_t)r * N_CH + c4);
    }

    // ---- async stage B: W[l,e] 128(K) x 128(N), 16 ops/thread ----
    const int l = l_of_row(r);
    const float* wb = w + ((size_t)(l * N_ELEM + e)) * N_CH * OUT_DIM;
#pragma unroll
    for (int i = tid; i < N_CH * (OUT_DIM / 4); i += 256) {
        int c  = i >> 5;
        int d4 = (i & 31) << 2;
        async_copy_b128(&Bs[c][d4], wb + c * OUT_DIM + d4);
    }

    wait_async_zero();   // this wave's async writes to LDS are complete
    __syncthreads();     // all waves' writes complete

    const int wave = tid >> 5;        // 0..7 -> output column tile
    const int lane = tid & 31;
    const int half = lane >> 4;       // K sub-block select per ISA A/B layout
    const int lr   = lane & 15;
    const int dcol = wave * 16 + lr;

#pragma unroll
    for (int s = 0; s < M_TILE / 16; ++s) {
        v8f acc = {0.f, 0.f, 0.f, 0.f, 0.f, 0.f, 0.f, 0.f};
#pragma unroll
        for (int kb = 0; kb < N_CH; kb += 4) {
            const int k = kb + 2 * half;            // even -> 8B-aligned b64
            // A 16x4 frag: lanes 0-15 M=lr,K={0,1}; lanes 16-31 M=lr,K={2,3}
            v2f a = *(const v2f*)&As[s * 16 + lr][k];
            // B 4x16 frag (mirror), natural layout: rows k and k+1, col dcol
            v2f b = { Bs[k][dcol], Bs[k + 1][dcol] };
            acc = __builtin_amdgcn_wmma_f32_16x16x4_f32(
                /*neg_a=*/false, a, /*neg_b=*/false, b,
                /*c_mod=*/(short)0, acc, /*reuse_a=*/false, /*reuse_b=*/false);
        }
        // C/D layout: VGPR j -> M = s*16 + j + 8*half, N = dcol
#pragma unroll
        for (int j = 0; j < 8; ++j) {
            int m = s * 16 + j + 8 * half;
            if (m0 + m < cnt) {
                out[(size_t)nids[m] * (N_ROWS * OUT_DIM) + (size_t)r * OUT_DIM + dcol]
                    = acc[j] * PATH_W;
            }
        }
    }
}

extern "C" void kernel_launch(void* const* d_in, const int* in_sizes, int n_in,
                              void* d_out, int out_size, void* d_ws, size_t ws_size,
                              hipStream_t stream) {
    (void)in_sizes; (void)n_in; (void)out_size; (void)ws_size;
    const float* x  = (const float*)d_in[0];
    const int*   y  = (const int*)d_in[1];
    const float* wt = (const float*)d_in[2];
    float* out = (float*)d_out;
    int* ws = (int*)d_ws;

    const int nblk = (N_NODES + 255) / 256;

    zero_ws_kernel<<<1, 64, 0, stream>>>(ws);
    count_kernel  <<<nblk, 256, 0, stream>>>(y, ws);
    scan_kernel   <<<1, 32, 0, stream>>>(ws);
    scatter_kernel<<<nblk, 256, 0, stream>>>(y, ws);

    dim3 grid((N_NODES + M_TILE - 1) / M_TILE, N_ROWS, N_ELEM);
    gemm_kernel<<<grid, 256, 0, stream>>>(x, wt, out, ws);
}